// MLP_84679575208563
// MI455X (gfx1250) — compile-verified
//
#include <hip/hip_runtime.h>
#include <math.h>

typedef float v2f __attribute__((ext_vector_type(2)));
typedef float v8f __attribute__((ext_vector_type(8)));

#define XSS 1028   // padded row stride for 64ch*16comp x-tile (bank stride 4)
#define GBSS 132   // padded row stride for 128-wide scalar tile

// ---------------- compile-time Pin(3,0,1) tables ----------------
namespace ga {
constexpr int popc(int x){ int c=0; for(int i=0;i<8;i++) c += (x>>i)&1; return c; }
constexpr int reorder_sign(int a,int b){
  int s=0; int aa=a>>1;
  while(aa){ s += popc(aa & b); aa >>= 1; }
  return (s&1) ? -1 : 1;
}
// blade order: 1,e0,e1,e2,e3,e01,e02,e03,e12,e13,e23,e012,e013,e023,e123,e0123
constexpr int MASK_OF[16] = {0,1,2,4,8,3,5,9,6,10,12,7,11,13,14,15};
constexpr int idx_of(int m){ int r=0; for(int i=0;i<16;i++) if(MASK_OF[i]==m) r=i; return r; }
struct Tables {
  signed char gpk[16][16], gps[16][16];
  signed char jnk[16][16], jns[16][16];
  signed char e0src[16], kg[16];
};
constexpr Tables make_tables(){
  Tables t{};
  for(int i=0;i<16;i++){
    int mi = MASK_OF[i];
    t.kg[i] = (signed char)popc(mi);
    t.e0src[i] = (mi & 1) ? (signed char)idx_of(mi & ~1) : (signed char)(-1);
    for(int j=0;j<16;j++){
      int mj = MASK_OF[j];
      if(mi & mj & 1){ t.gps[i][j]=0; t.gpk[i][j]=0; }           // e0^2 = 0
      else { t.gps[i][j]=(signed char)reorder_sign(mi,mj);
             t.gpk[i][j]=(signed char)idx_of(mi^mj); }
      int ci = 15^mi, cj = 15^mj;                                 // duals
      if(ci & cj){ t.jns[i][j]=0; t.jnk[i][j]=0; }
      else {
        int rm = ci|cj;
        int w  = reorder_sign(ci,cj);
        int si = reorder_sign(mi,ci);
        int sj = reorder_sign(mj,cj);
        int mk = 15^rm;
        int sk = reorder_sign(mk,rm);
        t.jns[i][j]=(signed char)(si*sj*sk*w);
        t.jnk[i][j]=(signed char)idx_of(mk);
      }
    }
  }
  return t;
}
constexpr Tables T = make_tables();
} // namespace ga

// ---------------- WMMA helpers ----------------
__device__ inline v8f vzero8(){ v8f z; 
#pragma unroll
  for(int i=0;i<8;i++) z[i]=0.f; return z; }

__device__ inline v8f wmma4(v2f a, v2f b, v8f c){
  return __builtin_amdgcn_wmma_f32_16x16x4_f32(false, a, false, b, (short)0, c, false, false);
}
// A frag: lane l -> row m=l&15, K half h=l>>4; v0=A[m][k0+2h], v1=A[m][k0+2h+1]
__device__ inline v2f lda_mv(const float* xs, int m, int hh, int k0, int comp){
  const float* p = xs + m*XSS + (k0 + 2*hh)*16 + comp;
  v2f a; a[0]=p[0]; a[1]=p[16]; return a;
}
__device__ inline v2f lda_s(const float* ss, int sst, int m, int hh, int k0){
  const float* p = ss + m*sst + k0 + 2*hh;
  v2f a; a[0]=p[0]; a[1]=p[1]; return a;
}
// B frag from [K][ldn] weight: col = col0 + (l&15)
__device__ inline v2f ldb(const float* w, int hh, int k0, int ldn, int col){
  const float* p = w + (size_t)(k0 + 2*hh)*ldn + col;
  v2f b; b[0]=p[0]; b[1]=p[ldn]; return b;
}
__device__ inline float gelu_t(float x){
  float x3 = x*x*x;
  return 0.5f*x*(1.0f + tanhf(0.7978845608028654f*(x + 0.044715f*x3)));
}

// ---------------- weight reorg ----------------
__global__ void reorg_w9(const float* __restrict__ src, float* __restrict__ dst, int Co, int Ci){
  int idx = blockIdx.x*256 + threadIdx.x;
  if(idx < Co*Ci*9){
    int o = idx / (Ci*9);
    int rem = idx - o*Ci*9;
    int c = rem / 9, g = rem - c*9;
    dst[((size_t)g*Ci + c)*Co + o] = src[idx];
  }
}
__global__ void transp(const float* __restrict__ src, float* __restrict__ dst, int R, int Cn){
  int idx = blockIdx.x*256 + threadIdx.x;
  if(idx < R*Cn){
    int r = idx / Cn, c = idx - r*Cn;
    dst[(size_t)c*R + r] = src[idx];
  }
}

// ---------------- kernel 1: fused 4x equi_linear + GP + join -> hidden ----------------
__global__ __launch_bounds__(128) void gb4_kernel(
    const float* __restrict__ xmv, const float* __restrict__ scal,
    const float* __restrict__ wr,   // [4][9][64][32]
    const float* __restrict__ s2mv, // [4][128][32]
    const float* __restrict__ bmv,  // [4*32]
    float* __restrict__ hidden)     // [N][64][16]
{
  extern __shared__ float smem[];
  float* xs = smem;            // [16][XSS]
  float* ss = smem + 16*XSS;   // [16][GBSS]
  int tid = threadIdx.x;
  int row0 = blockIdx.x * 16;
  {
    const float4* src = (const float4*)(xmv + (size_t)row0*1024);
#pragma unroll
    for(int i=0;i<32;i++){
      int t4 = tid + i*128, g = t4*4;
      *(float4*)(xs + (g>>10)*XSS + (g&1023)) = src[t4];
    }
    const float4* s4 = (const float4*)(scal + (size_t)row0*128);
#pragma unroll
    for(int i=0;i<4;i++){
      int t4 = tid + i*128, g = t4*4;
      *(float4*)(ss + (g>>7)*GBSS + (g&127)) = s4[t4];
    }
  }
  __syncthreads();

  int lane = tid & 31, wave = tid >> 5;
  int pair = wave >> 1, t = wave & 1;      // pair0: left/right(GP), pair1: lj/rj(join)
  int m = lane & 15, hh = lane >> 4;
  const float* WL = wr + (size_t)(pair*2+0)*9*64*32;
  const float* WR = wr + (size_t)(pair*2+1)*9*64*32;

  v8f accL[16], accR[16];
#pragma unroll
  for(int j=0;j<16;j++){ accL[j]=vzero8(); accR[j]=vzero8(); }

#pragma unroll
  for(int j=0;j<16;j++){
    const int kg = ga::T.kg[j];
    const float* wlg = WL + kg*2048;
    const float* wrg = WR + kg*2048;
    for(int k0=0;k0<64;k0+=4){
      v2f a  = lda_mv(xs,m,hh,k0,j);
      v2f bl = ldb(wlg,hh,k0,32,t*16+m);
      v2f br = ldb(wrg,hh,k0,32,t*16+m);
      accL[j]=wmma4(a,bl,accL[j]);
      accR[j]=wmma4(a,br,accR[j]);
    }
    const int src = ga::T.e0src[j];
    if(src >= 0){
      const float* wle = WL + (4+kg)*2048;
      const float* wre = WR + (4+kg)*2048;
      for(int k0=0;k0<64;k0+=4){
        v2f a  = lda_mv(xs,m,hh,k0,src);
        v2f bl = ldb(wle,hh,k0,32,t*16+m);
        v2f br = ldb(wre,hh,k0,32,t*16+m);
        accL[j]=wmma4(a,bl,accL[j]);
        accR[j]=wmma4(a,br,accR[j]);
      }
    }
  }
  // scalar -> component 0 (s @ w_s2mv.T), K = 128
  {
    const float* SL = s2mv + (size_t)(pair*2+0)*128*32;
    const float* SR = s2mv + (size_t)(pair*2+1)*128*32;
    for(int k0=0;k0<128;k0+=4){
      v2f a  = lda_s(ss,GBSS,m,hh,k0);
      v2f bl = ldb(SL,hh,k0,32,t*16+m);
      v2f br = ldb(SR,hh,k0,32,t*16+m);
      accL[0]=wmma4(a,bl,accL[0]);
      accR[0]=wmma4(a,br,accR[0]);
    }
    float bl_ = bmv[(pair*2+0)*32 + t*16 + m];
    float br_ = bmv[(pair*2+1)*32 + t*16 + m];
#pragma unroll
    for(int r=0;r<8;r++){ accL[0][r]+=bl_; accR[0][r]+=br_; }
  }
  // in-lane geometric product / join, write hidden
#pragma unroll
  for(int r=0;r<8;r++){
    float L[16], R[16], h[16];
#pragma unroll
    for(int j=0;j<16;j++){ L[j]=accL[j][r]; R[j]=accR[j][r]; h[j]=0.f; }
    if(pair==0){
#pragma unroll
      for(int i=0;i<16;i++)
#pragma unroll
        for(int j=0;j<16;j++){
          const int s = ga::T.gps[i][j];
          if(s) h[(int)ga::T.gpk[i][j]] += (float)s * L[i]*R[j];
        }
    } else {
#pragma unroll
      for(int i=0;i<16;i++)
#pragma unroll
        for(int j=0;j<16;j++){
          const int s = ga::T.jns[i][j];
          if(s) h[(int)ga::T.jnk[i][j]] += (float)s * L[i]*R[j];  // REF_PSS = 1
        }
    }
    int row = row0 + r + 8*hh;
    int ch  = pair*32 + t*16 + m;
    float* o = hidden + ((size_t)row*64 + ch)*16;
    *(float4*)(o+ 0) = make_float4(h[0],h[1],h[2],h[3]);
    *(float4*)(o+ 4) = make_float4(h[4],h[5],h[6],h[7]);
    *(float4*)(o+ 8) = make_float4(h[8],h[9],h[10],h[11]);
    *(float4*)(o+12) = make_float4(h[12],h[13],h[14],h[15]);
  }
}

// ---------------- kernel 2: equi_linear block (mv + scalar, opt norm+gelu) ----------------
__global__ __launch_bounds__(128) void equi_block_kernel(
    const float* __restrict__ xmv, const float* __restrict__ xscal,
    const float* __restrict__ wr9,    // [9][64][64]
    const float* __restrict__ s2mv,   // [Sin][64]
    const float* __restrict__ mvs2s,  // [64][Sout]
    const float* __restrict__ s2s,    // [Sin][Sout]
    const float* __restrict__ bmv, const float* __restrict__ bs,
    float* __restrict__ omv, float* __restrict__ os,
    int Sin, int Sout, int do_na)
{
  extern __shared__ float smem[];
  float* xs = smem;
  int sst = Sin + 4;
  float* ss  = smem + 16*XSS;
  float* red = ss + 16*sst;   // [48]: rowsq | ssum | ssumsq
  int tid = threadIdx.x;
  int row0 = blockIdx.x*16;
  {
    const float4* src = (const float4*)(xmv + (size_t)row0*1024);
#pragma unroll
    for(int i=0;i<32;i++){
      int t4 = tid + i*128, g = t4*4;
      *(float4*)(xs + (g>>10)*XSS + (g&1023)) = src[t4];
    }
    int nf4 = 16*Sin/4;
    const float4* s4 = (const float4*)(xscal + (size_t)row0*Sin);
    for(int t4=tid; t4<nf4; t4+=128){
      int g = t4*4, r = g/Sin, off = g - r*Sin;
      *(float4*)(ss + r*sst + off) = s4[t4];
    }
  }
  if(tid < 48) red[tid] = 0.f;
  __syncthreads();

  int lane = tid & 31, wave = tid >> 5, m = lane & 15, hh = lane >> 4;

  v8f acc[16];
#pragma unroll
  for(int j=0;j<16;j++) acc[j]=vzero8();

#pragma unroll
  for(int j=0;j<16;j++){
    const int kg = ga::T.kg[j];
    const float* w = wr9 + kg*4096;
    for(int k0=0;k0<64;k0+=4){
      v2f a = lda_mv(xs,m,hh,k0,j);
      v2f b = ldb(w,hh,k0,64,wave*16+m);
      acc[j]=wmma4(a,b,acc[j]);
    }
    const int sidx = ga::T.e0src[j];
    if(sidx >= 0){
      const float* we = wr9 + (4+kg)*4096;
      for(int k0=0;k0<64;k0+=4){
        v2f a = lda_mv(xs,m,hh,k0,sidx);
        v2f b = ldb(we,hh,k0,64,wave*16+m);
        acc[j]=wmma4(a,b,acc[j]);
      }
    }
  }
  for(int k0=0;k0<Sin;k0+=4){
    v2f a = lda_s(ss,sst,m,hh,k0);
    v2f b = ldb(s2mv,hh,k0,64,wave*16+m);
    acc[0]=wmma4(a,b,acc[0]);
  }
  {
    float bb = bmv[wave*16+m];
#pragma unroll
    for(int r=0;r<8;r++) acc[0][r]+=bb;
  }
  // scalar output GEMMs: s_out = x_mv[...,0] @ mvs2s + s @ s2s + b
  int ntpw = Sout >> 6;   // tiles of 16 per wave (2 or 4)
  v8f accS[4];
#pragma unroll
  for(int ti=0;ti<4;ti++) accS[ti]=vzero8();
#pragma unroll
  for(int ti=0;ti<4;ti++){
    if(ti < ntpw){
      int col0 = (wave + 4*ti)*16;
      for(int k0=0;k0<64;k0+=4){
        v2f a = lda_mv(xs,m,hh,k0,0);
        v2f b = ldb(mvs2s,hh,k0,Sout,col0+m);
        accS[ti]=wmma4(a,b,accS[ti]);
      }
      for(int k0=0;k0<Sin;k0+=4){
        v2f a = lda_s(ss,sst,m,hh,k0);
        v2f b = ldb(s2s,hh,k0,Sout,col0+m);
        accS[ti]=wmma4(a,b,accS[ti]);
      }
      float bb = bs[col0+m];
#pragma unroll
      for(int r=0;r<8;r++) accS[ti][r]+=bb;
    }
  }

  if(do_na){
#pragma unroll
    for(int r=0;r<8;r++){
      float sq = acc[0][r]*acc[0][r] + acc[2][r]*acc[2][r] + acc[3][r]*acc[3][r]
               + acc[4][r]*acc[4][r] + acc[8][r]*acc[8][r] + acc[9][r]*acc[9][r]
               + acc[10][r]*acc[10][r] + acc[14][r]*acc[14][r];
      int row = r + 8*hh;
      atomicAdd(&red[row], sq);
      float sum=0.f, sumsq=0.f;
#pragma unroll
      for(int ti=0;ti<4;ti++) if(ti<ntpw){ float v=accS[ti][r]; sum+=v; sumsq+=v*v; }
      atomicAdd(&red[16+row], sum);
      atomicAdd(&red[32+row], sumsq);
    }
    __syncthreads();
    if(tid < 16){
      float inv = 1.0f / sqrtf(fmaxf(red[tid]*(1.0f/64.0f), 0.01f));
      float mu  = red[16+tid] / (float)Sout;
      float var = red[32+tid] / (float)Sout - mu*mu;
      red[tid]    = inv;
      red[16+tid] = mu;
      red[32+tid] = 1.0f / sqrtf(var + 1e-5f);
    }
    __syncthreads();
#pragma unroll
    for(int r=0;r<8;r++){
      int row = r + 8*hh;
      float inv = red[row];
#pragma unroll
      for(int j=0;j<16;j++) acc[j][r] *= inv;
      float g = gelu_t(acc[0][r]);
#pragma unroll
      for(int j=0;j<16;j++) acc[j][r] *= g;
      float mu = red[16+row], rs = red[32+row];
#pragma unroll
      for(int ti=0;ti<4;ti++) if(ti<ntpw) accS[ti][r] = gelu_t((accS[ti][r]-mu)*rs);
    }
  }
#pragma unroll
  for(int r=0;r<8;r++){
    int row = row0 + r + 8*hh;
    int ch  = wave*16 + m;
    float* o = omv + ((size_t)row*64 + ch)*16;
    *(float4*)(o+ 0) = make_float4(acc[0][r],acc[1][r],acc[2][r],acc[3][r]);
    *(float4*)(o+ 4) = make_float4(acc[4][r],acc[5][r],acc[6][r],acc[7][r]);
    *(float4*)(o+ 8) = make_float4(acc[8][r],acc[9][r],acc[10][r],acc[11][r]);
    *(float4*)(o+12) = make_float4(acc[12][r],acc[13][r],acc[14][r],acc[15][r]);
#pragma unroll
    for(int ti=0;ti<4;ti++)
      if(ti<ntpw) os[(size_t)row*Sout + (wave+4*ti)*16 + m] = accS[ti][r];
  }
}

// ---------------- host ----------------
extern "C" void kernel_launch(void* const* d_in, const int* in_sizes, int n_in,
                              void* d_out, int out_size, void* d_ws, size_t ws_size,
                              hipStream_t stream)
{
  const float* x = (const float*)d_in[0];
  const float* s = (const float*)d_in[1];
  float* ws = (float*)d_ws;
  float* A  = ws;                 // hidden, later mv2   [N*64*16]
  float* B  = ws + 33554432;      // mv1
  float* S1 = ws + 67108864;      // [N*256]
  float* S2 = ws + 75497472;
  float* W  = ws + 83886080;
  float* wr_gb     = W;
  float* s2mv_gb   = W + 73728;
  float* wr_gbo    = W + 90112;
  float* s2mv_gbo  = W + 126976;
  float* mvs2s_gbo = W + 135168;
  float* s2s_gbo   = W + 151552;
  float* wr_l1     = W + 184320;
  float* s2mv_l1   = W + 221184;
  float* mvs2s_l1  = W + 237568;
  float* s2s_l1    = W + 253952;
  float* wr_l2     = W + 319488;
  float* s2mv_l2   = W + 356352;
  float* mvs2s_l2  = W + 372736;
  float* s2s_l2    = W + 380928;

  // weight reorganization (tiny, L2-resident afterwards)
  for(int li=0; li<4; ++li){
    reorg_w9<<<(32*64*9+255)/256,256,0,stream>>>((const float*)d_in[2] + li*32*64*9, wr_gb + li*9*64*32, 32, 64);
    transp  <<<(32*128 +255)/256,256,0,stream>>>((const float*)d_in[3] + li*32*128,  s2mv_gb + li*128*32, 32, 128);
  }
  reorg_w9<<<(64*64*9+255)/256,256,0,stream>>>((const float*)d_in[5],  wr_gbo, 64, 64);
  transp  <<<(64*128 +255)/256,256,0,stream>>>((const float*)d_in[6],  s2mv_gbo, 64, 128);
  transp  <<<(256*64 +255)/256,256,0,stream>>>((const float*)d_in[8],  mvs2s_gbo, 256, 64);
  transp  <<<(256*128+255)/256,256,0,stream>>>((const float*)d_in[9],  s2s_gbo, 256, 128);
  reorg_w9<<<(64*64*9+255)/256,256,0,stream>>>((const float*)d_in[11], wr_l1, 64, 64);
  transp  <<<(64*256 +255)/256,256,0,stream>>>((const float*)d_in[12], s2mv_l1, 64, 256);
  transp  <<<(256*64 +255)/256,256,0,stream>>>((const float*)d_in[14], mvs2s_l1, 256, 64);
  transp  <<<(256*256+255)/256,256,0,stream>>>((const float*)d_in[15], s2s_l1, 256, 256);
  reorg_w9<<<(64*64*9+255)/256,256,0,stream>>>((const float*)d_in[17], wr_l2, 64, 64);
  transp  <<<(64*256 +255)/256,256,0,stream>>>((const float*)d_in[18], s2mv_l2, 64, 256);
  transp  <<<(128*64 +255)/256,256,0,stream>>>((const float*)d_in[20], mvs2s_l2, 128, 64);
  transp  <<<(128*256+255)/256,256,0,stream>>>((const float*)d_in[21], s2s_l2, 128, 256);

  const int nblk = 32768/16;
  size_t sh_gb = (size_t)(16*XSS + 16*GBSS)*sizeof(float);
  gb4_kernel<<<nblk,128,sh_gb,stream>>>(x, s, wr_gb, s2mv_gb, (const float*)d_in[4], A);

  size_t sh1 = (size_t)(16*XSS + 16*(128+4) + 48)*sizeof(float);
  equi_block_kernel<<<nblk,128,sh1,stream>>>(A, s, wr_gbo, s2mv_gbo, mvs2s_gbo, s2s_gbo,
      (const float*)d_in[7], (const float*)d_in[10], B, S1, 128, 256, 1);

  size_t sh2 = (size_t)(16*XSS + 16*(256+4) + 48)*sizeof(float);
  equi_block_kernel<<<nblk,128,sh2,stream>>>(B, S1, wr_l1, s2mv_l1, mvs2s_l1, s2s_l1,
      (const float*)d_in[13], (const float*)d_in[16], A, S2, 256, 256, 1);

  float* out_mv = (float*)d_out;
  float* out_s  = (float*)d_out + 33554432;
  equi_block_kernel<<<nblk,128,sh2,stream>>>(A, S2, wr_l2, s2mv_l2, mvs2s_l2, s2s_l2,
      (const float*)d_in[19], (const float*)d_in[22], out_mv, out_s, 256, 128, 0);
}